// PhotonicMesh_69226282877482
// MI455X (gfx1250) — compile-verified
//
#include <hip/hip_runtime.h>
#include <hip/hip_bf16.h>

// CDNA5 / gfx1250 — wave32, WMMA 16x16x4 f32.
typedef __attribute__((ext_vector_type(2))) float v2f;
typedef __attribute__((ext_vector_type(8))) float v8f;

#define MESH_N 512
#define MESH_B 1024
#define ROWS_PER_BLK 4
// ATTEN = sqrt(10^(-0.2/10)) = 10^(-0.01)
#define MESH_ATTEN 0.97723722095581070f

// new = a*s + b*t, complex; c = (a.re, a.im, b.re, b.im)
__device__ __forceinline__ float2 pm_cmix(const float4 c, const float2 s, const float2 t)
{
    float2 o;
    o.x = c.x * s.x - c.y * s.y + c.z * t.x - c.w * t.y;
    o.y = c.x * s.y + c.y * s.x + c.z * t.y + c.w * t.x;
    return o;
}

// ---------------------------------------------------------------------------
// Phase 0: expand (theta, phi, role) into per-(layer,port) complex coeffs.
// ---------------------------------------------------------------------------
__global__ void pm_build_coeff(const float* __restrict__ thetas,
                               const float* __restrict__ phis,
                               const int* __restrict__ mzi_idx,
                               const int* __restrict__ role,
                               float4* __restrict__ coeff, int total)
{
    int i = blockIdx.x * blockDim.x + threadIdx.x;
    if (i >= total) return;
    int m = mzi_idx[i];
    int r = role[i];
    float th = thetas[m];
    float ct = cosf(th) * MESH_ATTEN;
    float st = sinf(th) * MESH_ATTEN;
    float4 c;
    if (r == 0) {            // upper port: a = e^{i phi} ct, b = e^{i phi} st
        float ph = phis[m];
        float cp = cosf(ph), sp = sinf(ph);
        c = make_float4(ct * cp, ct * sp, st * cp, st * sp);
    } else if (r == 1) {     // lower port: a = ct, b = -st
        c = make_float4(ct, 0.0f, -st, 0.0f);
    } else {                 // passthrough (partner == self): a = 1, b = 0
        c = make_float4(1.0f, 0.0f, 0.0f, 0.0f);
    }
    coeff[i] = c;
}

// ---------------------------------------------------------------------------
// Phase 1: evolve the 512x512 identity through all layers -> W = T0^T...TL^T.
// 256 threads/block; thread t OWNS port pair (2t, 2t+1) for ROWS_PER_BLK rows.
//   * even layers: pairs (2t,2t+1) -> pure register mixing, NO LDS/barrier
//   * odd layers : neighbor exchange via double-buffered LDS, 1 barrier
// Output written planar (Wr, Wi) with K-pairs packed:
//   Wr[(k/2)*N + n] = ( W[k][n].re, W[k+1][n].re )   -> direct b64 WMMA-B loads
// ---------------------------------------------------------------------------
__global__ __launch_bounds__(MESH_N / 2)
void pm_evolve_operator(const float4* __restrict__ coeff,
                        const int* __restrict__ partner,
                        float2* __restrict__ Wr, float2* __restrict__ Wi, int L)
{
    __shared__ float2 xsh[2][ROWS_PER_BLK][MESH_N];    // 32 KB
    const int t  = threadIdx.x;                        // 0..255
    const int p0 = 2 * t;
    const int p1 = 2 * t + 1;
    const int row0 = blockIdx.x * ROWS_PER_BLK;

    float2 slo[ROWS_PER_BLK], shi[ROWS_PER_BLK];
#pragma unroll
    for (int r = 0; r < ROWS_PER_BLK; ++r) {
        slo[r] = make_float2((p0 == row0 + r) ? 1.0f : 0.0f, 0.0f);
        shi[r] = make_float2((p1 == row0 + r) ? 1.0f : 0.0f, 0.0f);
    }

    for (int l = 0; l < L; l += 2) {
        // ---- even layer: pairs (2t, 2t+1) are thread-local ----
        {
            const float4 clo = coeff[l * MESH_N + p0];
            const float4 chi = coeff[l * MESH_N + p1];
#pragma unroll
            for (int r = 0; r < ROWS_PER_BLK; ++r) {
                const float2 a = slo[r], b = shi[r];
                slo[r] = pm_cmix(clo, a, b);
                shi[r] = pm_cmix(chi, b, a);
            }
        }
        // ---- odd layer: pairs (2t+1, 2t+2) cross threads ----
        {
            const int lo = l + 1;
            const float4 clo = coeff[lo * MESH_N + p0];
            const float4 chi = coeff[lo * MESH_N + p1];
            const int qlo = partner[lo * MESH_N + p0]; // p0-1 or p0 (edge)
            const int qhi = partner[lo * MESH_N + p1]; // p1+1 or p1 (edge)
            const int buf = (lo >> 1) & 1;             // alternates per odd layer
#pragma unroll
            for (int r = 0; r < ROWS_PER_BLK; ++r) {
                xsh[buf][r][p0] = slo[r];
                xsh[buf][r][p1] = shi[r];
            }
            __syncthreads();
#pragma unroll
            for (int r = 0; r < ROWS_PER_BLK; ++r) {
                const float2 tlo = xsh[buf][r][qlo];
                const float2 thi = xsh[buf][r][qhi];
                slo[r] = pm_cmix(clo, slo[r], tlo);
                shi[r] = pm_cmix(chi, shi[r], thi);
            }
            // Buffer `buf` is rewritten two odd-layers later, i.e. after the
            // next barrier -> every reader has finished. No second barrier.
        }
    }

    // Store planar real/imag with packed K-pairs (row0 is a multiple of 4).
#pragma unroll
    for (int rp = 0; rp < ROWS_PER_BLK; rp += 2) {
        const size_t kb = (size_t)((row0 + rp) >> 1) * MESH_N;
        Wr[kb + p0] = make_float2(slo[rp].x, slo[rp + 1].x);
        Wr[kb + p1] = make_float2(shi[rp].x, shi[rp + 1].x);
        Wi[kb + p0] = make_float2(slo[rp].y, slo[rp + 1].y);
        Wi[kb + p1] = make_float2(shi[rp].y, shi[rp + 1].y);
    }
}

// ---------------------------------------------------------------------------
// Phase 2: Y = X * W (X real 1024x512, W complex), out = |Y|^2.
// One wave computes a 16x32 tile: 4 f32 WMMA accumulators (real/imag x 2 N
// tiles) sharing each A load. B operands are single b64 loads (packed pairs).
// ---------------------------------------------------------------------------
__global__ __launch_bounds__(256)
void pm_gemm_photodet(const float* __restrict__ X,
                      const float2* __restrict__ Wr,
                      const float2* __restrict__ Wi,
                      float* __restrict__ out)
{
    const int lane = threadIdx.x & 31;
    const int wave = threadIdx.x >> 5;
    const int tile = blockIdx.x * 8 + wave;            // 0 .. 1023
    const int mt = tile >> 4;                          // 0 .. 63 (16-row tiles)
    const int nt = tile & 15;                          // 0 .. 15 (32-col tiles)
    const int m0 = mt * 16;
    const int n0 = nt * 32;

    const int half = lane >> 4;                        // lane group 0/1
    const int mrow = m0 + (lane & 15);                 // A: M = lane%16
    const int ncA  = n0 + (lane & 15);                 // first N tile column
    const int ncB  = ncA + 16;                         // second N tile column

    const float* __restrict__ xrow = X + (size_t)mrow * MESH_N;

    v8f aR0 = {}, aI0 = {}, aR1 = {}, aI1 = {};

#pragma unroll 4
    for (int k0 = 0; k0 < MESH_N; k0 += 4) {
        const int ka = k0 + 2 * half;                  // this half's K pair
        // A 16x4: lane holds X[M][ka], X[M][ka+1] -> one b64 load
        v2f a = *(const v2f*)(xrow + ka);
        // B 4x16: packed K-pair layout -> one b64 load per operand, no shuffles
        const size_t kb = (size_t)(ka >> 1) * MESH_N;
        v2f brA = *(const v2f*)(Wr + kb + ncA);
        v2f biA = *(const v2f*)(Wi + kb + ncA);
        v2f brB = *(const v2f*)(Wr + kb + ncB);
        v2f biB = *(const v2f*)(Wi + kb + ncB);
        aR0 = __builtin_amdgcn_wmma_f32_16x16x4_f32(false, a, false, brA, (short)0, aR0, false, false);
        aI0 = __builtin_amdgcn_wmma_f32_16x16x4_f32(false, a, false, biA, (short)0, aI0, false, false);
        aR1 = __builtin_amdgcn_wmma_f32_16x16x4_f32(false, a, false, brB, (short)0, aR1, false, false);
        aI1 = __builtin_amdgcn_wmma_f32_16x16x4_f32(false, a, false, biB, (short)0, aI1, false, false);
    }

    // D 16x16 f32: VGPR r -> row m0 + r + 8*half, col = lane%16 in each tile.
#pragma unroll
    for (int r = 0; r < 8; ++r) {
        const int m = m0 + r + 8 * half;
        out[(size_t)m * MESH_N + ncA] = aR0[r] * aR0[r] + aI0[r] * aI0[r];
        out[(size_t)m * MESH_N + ncB] = aR1[r] * aR1[r] + aI1[r] * aI1[r];
    }
}

// ---------------------------------------------------------------------------
// Launcher. Inputs (reference order):
//   0: x [1024*512] f32, 1: thetas f32, 2: phis f32,
//   3: partner [L*512] i32, 4: mzi_idx i32, 5: role i32.
// Workspace: [coeff float4 L*N | Wr float2 (N/2)*N | Wi float2 (N/2)*N] ~6 MB.
// ---------------------------------------------------------------------------
extern "C" void kernel_launch(void* const* d_in, const int* in_sizes, int n_in,
                              void* d_out, int out_size, void* d_ws, size_t ws_size,
                              hipStream_t stream)
{
    const float* x       = (const float*)d_in[0];
    const float* thetas  = (const float*)d_in[1];
    const float* phis    = (const float*)d_in[2];
    const int*   partner = (const int*)d_in[3];
    const int*   mzi_idx = (const int*)d_in[4];
    const int*   role    = (const int*)d_in[5];
    float* out = (float*)d_out;

    const int LN = in_sizes[3];        // L * N = 262144
    const int L  = LN / MESH_N;        // 512

    float4* coeff = (float4*)d_ws;
    float2* Wr = (float2*)((char*)d_ws + (size_t)LN * sizeof(float4));
    float2* Wi = Wr + (size_t)(MESH_N / 2) * MESH_N;

    // Phase 0: coefficient table
    pm_build_coeff<<<(LN + 255) / 256, 256, 0, stream>>>(
        thetas, phis, mzi_idx, role, coeff, LN);

    // Phase 1: evolve identity -> packed planar W (128 blocks x 256 threads)
    pm_evolve_operator<<<MESH_N / ROWS_PER_BLK, MESH_N / 2, 0, stream>>>(
        coeff, partner, Wr, Wi, L);

    // Phase 2: WMMA GEMM + fused |.|^2
    // 64 M-tiles * 16 N-pair-tiles = 1024 wave jobs, 8 waves/block -> 128 blocks
    pm_gemm_photodet<<<128, 256, 0, stream>>>(x, Wr, Wi, out);
}